// RefractiveSurface_86260123173029
// MI455X (gfx1250) — compile-verified
//
#include <hip/hip_runtime.h>
#include <hip/hip_bf16.h>

// Ray refraction through a spherical surface — pure streaming workload.
// 88 B/ray, ~60 FLOP/ray -> HBM-bound (floor ~16us @ 23.3 TB/s).
// Strategy: 4 rays/thread, all global traffic as coalesced b128;
// wavelengths staged via CDNA5 async global->LDS (ASYNCcnt) path.

#define RADIUS_F    50.0f
#define CENTER_Z_F  60.0f
#define CAUCHY_A2_F 1.5046f
#define CAUCHY_B2_F 0.0042f

union Pack12 { float4 q[3]; float f[12]; };
union Pack4  { float4 q;    float f[4];  };

__global__ __launch_bounds__(256)
void RefractiveSurface_kernel(const float* __restrict__ P,
                              const float* __restrict__ V,
                              const float* __restrict__ B,
                              const float* __restrict__ W,
                              float* __restrict__ out,
                              int nquads, int n)
{
    __shared__ float4 swl[256];
    const int tid = threadIdx.x;
    const int q   = blockIdx.x * 256 + tid;
    if (q >= nquads) return;

    // --- CDNA5 async data mover: stage 4 wavelengths (16B/lane) into LDS ---
    {
        unsigned lds_off = (unsigned)(uintptr_t)(&swl[tid]);            // wave-relative LDS byte offset
        unsigned long long gaddr =
            (unsigned long long)(uintptr_t)(W + (size_t)q * 4);
        asm volatile("global_load_async_to_lds_b128 %0, %1, off"
                     :: "v"(lds_off), "v"(gaddr) : "memory");
    }

    // --- Direct coalesced b128 loads for the 3-vector streams ---
    const float4* __restrict__ P4 = (const float4*)P;
    const float4* __restrict__ V4 = (const float4*)V;
    const float4* __restrict__ B4 = (const float4*)B;

    Pack12 p, v, rb;
    const size_t base = (size_t)3 * (size_t)q;
    p.q[0]  = P4[base + 0]; p.q[1]  = P4[base + 1]; p.q[2]  = P4[base + 2];
    v.q[0]  = V4[base + 0]; v.q[1]  = V4[base + 1]; v.q[2]  = V4[base + 2];
    rb.q[0] = B4[base + 0]; rb.q[1] = B4[base + 1]; rb.q[2] = B4[base + 2];

    // --- Wait for async DMA, pull wavelengths from LDS ---
    asm volatile("s_wait_asynccnt 0x0" ::: "memory");
    Pack4 w;
    w.q = swl[tid];

    Pack12 oP, oV, oB;
    Pack4  oW, oBV, oVR;

    #pragma unroll
    for (int r = 0; r < 4; ++r) {
        const float px = p.f[3*r+0], py = p.f[3*r+1], pz = p.f[3*r+2];
        const float vx = v.f[3*r+0], vy = v.f[3*r+1], vz = v.f[3*r+2];

        // Sphere intersection (|V| == 1 by construction)
        const float ocx = px, ocy = py, ocz = pz - CENTER_Z_F;
        const float bq  = ocx*vx + ocy*vy + ocz*vz;
        const float cq  = ocx*ocx + ocy*ocy + ocz*ocz - RADIUS_F*RADIUS_F;
        const float disc = bq*bq - cq;
        const bool  valid = (disc >= 0.0f);
        const float sq = valid ? sqrtf(disc) : 0.0f;          // guarded like the reference
        const float t  = -bq - sq;

        const float Xx = px + t*vx, Xy = py + t*vy, Xz = pz + t*vz;

        const float invR = 1.0f / RADIUS_F;
        const float nx = Xx*invR, ny = Xy*invR, nz = (Xz - CENTER_Z_F)*invR;

        const float dvn = vx*nx + vy*ny + vz*nz;
        const float s   = (dvn > 0.0f) ? -1.0f : 1.0f;        // nf = s * normals
        const float nfx = s*nx, nfy = s*ny, nfz = s*nz;
        const float cos_i = -(vx*nfx + vy*nfy + vz*nfz);

        // Cauchy dispersion: n1 = 1.0 exactly (B1 == 0)
        const float wl     = w.f[r];
        const float inv_l2 = 1.0f / (wl * wl);
        const float n2     = CAUCHY_A2_F + CAUCHY_B2_F * inv_l2;
        const float eta    = 1.0f / n2;

        const float sin2 = eta*eta * (1.0f - cos_i*cos_i);
        const bool  tir  = (sin2 > 1.0f);
        const float cos_t = tir ? 0.0f : sqrtf(1.0f - sin2);  // guarded like the reference

        const float k   = eta*cos_i - cos_t;
        const float rfx = eta*vx + k*nfx;                     // refracted
        const float rfy = eta*vy + k*nfy;
        const float rfz = eta*vz + k*nfz;
        const float c2  = 2.0f * cos_i;
        const float rlx = vx + c2*nfx;                        // reflected
        const float rly = vy + c2*nfy;
        const float rlz = vz + c2*nfz;

        const float ox = tir ? rlx : rfx;
        const float oy = tir ? rly : rfy;
        const float oz = tir ? rlz : rfz;

        const bool both = valid && !tir;

        oP.f[3*r+0] = both ? Xx : 0.0f;
        oP.f[3*r+1] = both ? Xy : 0.0f;
        oP.f[3*r+2] = both ? Xz : 0.0f;
        oV.f[3*r+0] = both ? ox : 0.0f;
        oV.f[3*r+1] = both ? oy : 0.0f;
        oV.f[3*r+2] = both ? oz : 0.0f;
        oB.f[3*r+0] = both ? rb.f[3*r+0] : 0.0f;
        oB.f[3*r+1] = both ? rb.f[3*r+1] : 0.0f;
        oB.f[3*r+2] = both ? rb.f[3*r+2] : 0.0f;
        oW.f[r]  = both ? wl : 0.0f;
        oBV.f[r] = both  ? 1.0f : 0.0f;
        oVR.f[r] = (!tir) ? 1.0f : 0.0f;
    }

    // --- Coalesced b128 stores: outputs concatenated in return order ---
    float4* __restrict__ outP  = (float4*)(out);
    float4* __restrict__ outV  = (float4*)(out + (size_t)3  * n);
    float4* __restrict__ outB  = (float4*)(out + (size_t)6  * n);
    float4* __restrict__ outW  = (float4*)(out + (size_t)9  * n);
    float4* __restrict__ outBV = (float4*)(out + (size_t)10 * n);
    float4* __restrict__ outVR = (float4*)(out + (size_t)11 * n);

    outP[base + 0] = oP.q[0]; outP[base + 1] = oP.q[1]; outP[base + 2] = oP.q[2];
    outV[base + 0] = oV.q[0]; outV[base + 1] = oV.q[1]; outV[base + 2] = oV.q[2];
    outB[base + 0] = oB.q[0]; outB[base + 1] = oB.q[1]; outB[base + 2] = oB.q[2];
    outW[q]  = oW.q;
    outBV[q] = oBV.q;
    outVR[q] = oVR.q;
}

extern "C" void kernel_launch(void* const* d_in, const int* in_sizes, int n_in,
                              void* d_out, int out_size, void* d_ws, size_t ws_size,
                              hipStream_t stream) {
    const float* P = (const float*)d_in[0];   // (N,3)
    const float* V = (const float*)d_in[1];   // (N,3)
    const float* B = (const float*)d_in[2];   // (N,3)
    const float* W = (const float*)d_in[3];   // (N,)

    const int n      = in_sizes[3];           // N_RAYS
    const int nquads = n / 4;                 // N_RAYS % 4 == 0

    dim3 block(256);
    dim3 grid((nquads + 255) / 256);
    RefractiveSurface_kernel<<<grid, block, 0, stream>>>(P, V, B, W,
                                                         (float*)d_out, nquads, n);
}